// Rank1HardTention_29326036697692
// MI455X (gfx1250) — compile-verified
//
#include <hip/hip_runtime.h>
#include <stdint.h>

#define B_ 2
#define S_ 4096
#define D_ 64
#define H_ 8

typedef float v2f __attribute__((ext_vector_type(2)));
typedef float v4f __attribute__((ext_vector_type(4)));
typedef float v8f __attribute__((ext_vector_type(8)));

// Monotonic float -> uint mapping (total order, matches float compare)
__device__ __forceinline__ uint32_t ord_f32(float v) {
    uint32_t b = __float_as_uint(v);
    return (b & 0x80000000u) ? ~b : (b | 0x80000000u);
}

// ---------------------------------------------------------------------------
// Kernel 0: initialize the 32 u64 reduction keys (16 max-keys, 16 min-keys)
// ---------------------------------------------------------------------------
__global__ void k_init(unsigned long long* __restrict__ keys) {
    int i = threadIdx.x;
    if (i < B_ * H_)            keys[i] = 0ull;   // running max keys
    else if (i < 2 * B_ * H_)   keys[i] = ~0ull;  // running min keys
}

// ---------------------------------------------------------------------------
// Kernel 1: k[b,h,t] = x[b,t,:] . ks[h,:] via V_WMMA_F32_16X16X4_F32,
// then global argmax/argmin per (b,h) with first-occurrence tie-breaking
// encoded into u64 atomic max/min keys.
// Grid: B_*32 blocks x 256 threads (8 waves); each wave owns one 16-row tile.
// ---------------------------------------------------------------------------
__global__ void __launch_bounds__(256)
k_select(const float* __restrict__ x, const float* __restrict__ ks,
         unsigned long long* __restrict__ keys) {
    const int b    = blockIdx.x >> 5;          // 32 blocks per batch
    const int part = blockIdx.x & 31;
    const int wave = threadIdx.x >> 5;
    const int lane = threadIdx.x & 31;
    const int tile = part * 8 + wave;          // 0..255
    const int t0   = tile * 16;

    const int n    = lane & 15;                // A: row, B: column (head)
    const int koff = (lane >> 4) * 2;          // K-offset from lane half

    // B = ks^T, zero-padded to 16 columns; 16 K-steps of 4, preloaded.
    v2f breg[16];
#pragma unroll
    for (int s = 0; s < 16; ++s) {
        int K = s * 4 + koff;
        if (n < H_) { breg[s].x = ks[n * D_ + K]; breg[s].y = ks[n * D_ + K + 1]; }
        else        { breg[s].x = 0.f;            breg[s].y = 0.f; }
    }

    const float* arow = x + ((size_t)b * S_ + (size_t)(t0 + n)) * D_;
    v8f c = {};
#pragma unroll
    for (int s = 0; s < 16; ++s) {
        int K = s * 4 + koff;
        v2f a;
        a.x = arow[K];
        a.y = arow[K + 1];
        c = __builtin_amdgcn_wmma_f32_16x16x4_f32(false, a, false, breg[s],
                                                  (short)0, c, false, false);
    }

    // lane holds c[i] = k[t0 + i + 8*(lane>>4)][head n]  (n < 8 valid)
    if (n < H_) {
        float mx = -__builtin_huge_valf(); int mxi = 0;
        float mn =  __builtin_huge_valf(); int mni = 0;
        const int tb = t0 + 8 * (lane >> 4);
#pragma unroll
        for (int i = 0; i < 8; ++i) {          // ascending t -> strict compares
            float v = c[i];                    // give first-occurrence ties
            if (v > mx) { mx = v; mxi = tb + i; }
            if (v < mn) { mn = v; mni = tb + i; }
        }
        unsigned long long kmax =
            ((unsigned long long)ord_f32(mx) << 32) |
            (unsigned long long)(0xFFFFFFFFu - (uint32_t)mxi);
        unsigned long long kmin =
            ((unsigned long long)ord_f32(mn) << 32) |
            (unsigned long long)(uint32_t)mni;
        atomicMax(&keys[b * H_ + n], kmax);
        atomicMin(&keys[B_ * H_ + b * H_ + n], kmin);
    }
}

// ---------------------------------------------------------------------------
// Kernel 2: q[b,h,s] via WMMA; select candidate row per (s,h) by sign(q);
// sum over heads, normalize, store. One wave per 16 queries.
// ---------------------------------------------------------------------------
__global__ void __launch_bounds__(32)
k_output(const float* __restrict__ x, const float* __restrict__ qs,
         const unsigned long long* __restrict__ keys,
         float* __restrict__ out) {
    __shared__ __align__(16) float cand[H_][2][D_];   // [head][max/min][d]
    __shared__ __align__(16) float row0[D_];          // q==0 edge case row
    __shared__ float qv[16][16];
    __shared__ float partial[16][2];
    __shared__ int   rowsel[H_][2];

    const int b    = blockIdx.y;
    const int s0   = blockIdx.x * 16;
    const int lane = threadIdx.x;
    const int n    = lane & 15;
    const int koff = (lane >> 4) * 2;

    if (lane < H_) {
        unsigned long long kmax = keys[b * H_ + lane];
        unsigned long long kmin = keys[B_ * H_ + b * H_ + lane];
        rowsel[lane][0] = (int)(0xFFFFFFFFu - (uint32_t)(kmax & 0xFFFFFFFFull));
        rowsel[lane][1] = (int)(uint32_t)(kmin & 0xFFFFFFFFull);
    }
    __syncthreads();

    // Stage candidate rows (hot in L2) into LDS: 8*2*64 + 64 floats.
    for (int u = lane; u < H_ * 2 * D_ + D_; u += 32) {
        if (u < H_ * 2 * D_) {
            int h = u >> 7, which = (u >> 6) & 1, d = u & 63;
            cand[h][which][d] =
                x[((size_t)b * S_ + (size_t)rowsel[h][which]) * D_ + d];
        } else {
            int d = u - H_ * 2 * D_;
            row0[d] = x[(size_t)b * S_ * D_ + d];
        }
    }

    // Q = X_tile x qs^T via WMMA (same layout as kernel 1)
    v2f breg[16];
#pragma unroll
    for (int s = 0; s < 16; ++s) {
        int K = s * 4 + koff;
        if (n < H_) { breg[s].x = qs[n * D_ + K]; breg[s].y = qs[n * D_ + K + 1]; }
        else        { breg[s].x = 0.f;            breg[s].y = 0.f; }
    }
    const float* arow = x + ((size_t)b * S_ + (size_t)(s0 + n)) * D_;
    v8f c = {};
#pragma unroll
    for (int s = 0; s < 16; ++s) {
        int K = s * 4 + koff;
        v2f a;
        a.x = arow[K];
        a.y = arow[K + 1];
        c = __builtin_amdgcn_wmma_f32_16x16x4_f32(false, a, false, breg[s],
                                                  (short)0, c, false, false);
    }
#pragma unroll
    for (int i = 0; i < 8; ++i)
        qv[i + 8 * (lane >> 4)][n] = c[i];
    __syncthreads();

    // Each lane: 32 output floats of one query row (sl = s-local, d0 = half)
    const int sl = lane >> 1;
    const int d0 = (lane & 1) * 32;
    v4f acc[8];
#pragma unroll
    for (int j = 0; j < 8; ++j) { v4f z = {0.f, 0.f, 0.f, 0.f}; acc[j] = z; }

#pragma unroll
    for (int h = 0; h < H_; ++h) {
        float q = qv[sl][h];
        const float* src = (q > 0.f) ? &cand[h][0][0]
                         : (q < 0.f) ? &cand[h][1][0]
                                     : &row0[0];
#pragma unroll
        for (int j = 0; j < 8; ++j)
            acc[j] += *(const v4f*)(src + d0 + j * 4);
    }

    float ss = 0.f;
#pragma unroll
    for (int j = 0; j < 8; ++j)
        ss += acc[j].x * acc[j].x + acc[j].y * acc[j].y +
              acc[j].z * acc[j].z + acc[j].w * acc[j].w;
    partial[sl][lane & 1] = ss;
    __syncthreads();

    const float inv = 1.0f / __builtin_sqrtf(partial[sl][0] + partial[sl][1]);
    float* orow = out + ((size_t)b * S_ + (size_t)(s0 + sl)) * D_ + d0;
#pragma unroll
    for (int j = 0; j < 8; ++j) {
        v4f v = acc[j] * inv;
        *(v4f*)(orow + j * 4) = v;
    }
}

// ---------------------------------------------------------------------------
extern "C" void kernel_launch(void* const* d_in, const int* in_sizes, int n_in,
                              void* d_out, int out_size, void* d_ws, size_t ws_size,
                              hipStream_t stream) {
    (void)in_sizes; (void)n_in; (void)out_size; (void)ws_size;
    const float* x  = (const float*)d_in[0];   // [B,S,D] f32
    const float* qs = (const float*)d_in[1];   // [H,1,D] f32
    const float* ks = (const float*)d_in[2];   // [H,1,D] f32
    float* out = (float*)d_out;                // [B,S,D] f32
    unsigned long long* keys = (unsigned long long*)d_ws; // 32 u64

    k_init<<<1, 32, 0, stream>>>(keys);
    k_select<<<B_ * 32, 256, 0, stream>>>(x, ks, keys);
    dim3 g2(S_ / 16, B_);
    k_output<<<g2, 32, 0, stream>>>(x, qs, keys, out);
}